// Model_46377056862708
// MI455X (gfx1250) — compile-verified
//
#include <hip/hip_runtime.h>
#include <hip/hip_bf16.h>

// ---------------------------------------------------------------------------
// Model dims (compile-time constants from the reference)
// ---------------------------------------------------------------------------
#define BATCH     32
#define SEQ_L     512
#define MVAR      32
#define PRED      96
#define P_LEN     16
#define STRIDE    8
#define N_PATCH   63                  // (512-16)/8+1
#define D_MODEL   128
#define D_STATE   16
#define D_CONV    4
#define D_INNER   256
#define DT_RANK   8
#define D_FF      256
#define E_LAYERS  2
#define NBM       (BATCH * MVAR)      // 1024 sequences
#define TOK       (NBM * N_PATCH)     // 64512 tokens
#define HEAD_K    (N_PATCH * D_MODEL) // 8064
#define SPLITK    8
#define HEAD_KC   (HEAD_K / SPLITK)   // 1008 (multiple of 4)

typedef float v2f __attribute__((ext_vector_type(2)));
typedef float v8f __attribute__((ext_vector_type(8)));

// Scheduling fence: keep prefetch loads issued BEFORE the WMMA block of the
// current pipeline stage (prevents the allocator/scheduler from recycling an
// operand register and re-issuing its load right before the consuming WMMA).
#if defined(__has_builtin)
#  if __has_builtin(__builtin_amdgcn_sched_barrier)
#    define SCHED_FENCE() __builtin_amdgcn_sched_barrier(0)
#  endif
#endif
#ifndef SCHED_FENCE
#  define SCHED_FENCE()
#endif

// Epilogue modes for the generic WMMA GEMM
#define EPI_STORE      0
#define EPI_BIAS_GELU  1
#define EPI_ACC        2
#define EPI_BIAS_ADD   3

#define WMMA_F32(a, b, c) \
    __builtin_amdgcn_wmma_f32_16x16x4_f32(false, (a), false, (b), (short)0, (c), false, false)

// ---------------------------------------------------------------------------
// Generic GEMM via V_WMMA_F32_16X16X4_F32:  C[M,N] = A[M,K] * W[N,K]^T
// Each wave computes TWO 16-row M tiles x FOUR 16-col N tiles (32x64 block):
// per k-step, 2 A fragments + 4 B fragments feed 8 WMMAs into 8 independent
// accumulator chains (~10.7 flop/byte from cache, good latency hiding).
// Operand loads are double-buffered (prefetch k+4 while computing k) in
// distinct named registers, with a sched_barrier so ALL six next-step loads
// issue before the current eight WMMAs (full software pipeline).
// Out-of-range W rows are CLAMPED to row N-1 (their products land only in
// D columns >= N, which are never stored) -> no divergent loads, EXEC stays
// all-ones through the whole K loop as WMMA requires.
// Per-lane operand layout (A 16x4 and its mirrored B 4x16):
//   row = lane&15, kh = (lane>>4)*2 ; v = {X[row, k+kh], X[row, k+kh+1]}
// D layout: vgpr r -> row (r + 8*(lane>=16)), col = lane&15.
// ---------------------------------------------------------------------------
template <int EPI>
__global__ __launch_bounds__(128) void gemm_wmma_f32(
    const float* __restrict__ A, int lda,
    const float* __restrict__ W, int ldw,
    float* __restrict__ C, int ldc,
    int Mrows, int N, int K,
    const float* __restrict__ bias,
    const float* __restrict__ R)
{
    const int lane    = threadIdx.x & 31;
    const int wave    = threadIdx.x >> 5;
    const int tilesN  = (N + 15) >> 4;
    const int groupsN = (tilesN + 3) >> 2;      // 4 N-tiles per wave
    const int groupsM = Mrows >> 5;             // 2 M-tiles per wave (Mrows % 32 == 0)
    const int gid     = blockIdx.x * 4 + wave;
    if (gid >= groupsM * groupsN) return;       // wave-uniform: EXEC stays all-ones

    const int tm  = (gid / groupsN) << 5;       // 32 rows
    const int tn0 = (gid % groupsN) << 6;       // 64 columns
    const int row = lane & 15;
    const int kh  = (lane >> 4) << 1;           // 0 or 2

    const float* Ar0 = A + (size_t)(tm + row) * lda + kh;
    const float* Ar1 = Ar0 + (size_t)16 * lda;
    const float* Wr0;
    const float* Wr1;
    const float* Wr2;
    const float* Wr3;
    {
        int w0 = tn0 + 0  + row; w0 = (w0 < N) ? w0 : (N - 1);
        int w1 = tn0 + 16 + row; w1 = (w1 < N) ? w1 : (N - 1);
        int w2 = tn0 + 32 + row; w2 = (w2 < N) ? w2 : (N - 1);
        int w3 = tn0 + 48 + row; w3 = (w3 < N) ? w3 : (N - 1);
        Wr0 = W + (size_t)w0 * ldw + kh;
        Wr1 = W + (size_t)w1 * ldw + kh;
        Wr2 = W + (size_t)w2 * ldw + kh;
        Wr3 = W + (size_t)w3 * ldw + kh;
    }

    v8f dA[4], dB[4];
#pragma unroll
    for (int t = 0; t < 4; ++t) { dA[t] = (v8f){}; dB[t] = (v8f){}; }

    // current operands (k = 0)
    v2f a0 = *(const v2f*)(Ar0);
    v2f a1 = *(const v2f*)(Ar1);
    v2f b0 = *(const v2f*)(Wr0);
    v2f b1 = *(const v2f*)(Wr1);
    v2f b2 = *(const v2f*)(Wr2);
    v2f b3 = *(const v2f*)(Wr3);

    for (int k = 0; k < K - 4; k += 4) {        // K is always a multiple of 4
        // prefetch next k-step into distinct registers
        const v2f na0 = *(const v2f*)(Ar0 + k + 4);
        const v2f na1 = *(const v2f*)(Ar1 + k + 4);
        const v2f nb0 = *(const v2f*)(Wr0 + k + 4);
        const v2f nb1 = *(const v2f*)(Wr1 + k + 4);
        const v2f nb2 = *(const v2f*)(Wr2 + k + 4);
        const v2f nb3 = *(const v2f*)(Wr3 + k + 4);

        SCHED_FENCE();   // all 6 prefetch loads issue before the 8 WMMAs below

        dA[0] = WMMA_F32(a0, b0, dA[0]);
        dB[0] = WMMA_F32(a1, b0, dB[0]);
        dA[1] = WMMA_F32(a0, b1, dA[1]);
        dB[1] = WMMA_F32(a1, b1, dB[1]);
        dA[2] = WMMA_F32(a0, b2, dA[2]);
        dB[2] = WMMA_F32(a1, b2, dB[2]);
        dA[3] = WMMA_F32(a0, b3, dA[3]);
        dB[3] = WMMA_F32(a1, b3, dB[3]);

        a0 = na0; a1 = na1; b0 = nb0; b1 = nb1; b2 = nb2; b3 = nb3;
    }
    // tail k-step
    dA[0] = WMMA_F32(a0, b0, dA[0]);
    dB[0] = WMMA_F32(a1, b0, dB[0]);
    dA[1] = WMMA_F32(a0, b1, dA[1]);
    dB[1] = WMMA_F32(a1, b1, dB[1]);
    dA[2] = WMMA_F32(a0, b2, dA[2]);
    dB[2] = WMMA_F32(a1, b2, dB[2]);
    dA[3] = WMMA_F32(a0, b3, dA[3]);
    dB[3] = WMMA_F32(a1, b3, dB[3]);

    const int rbase = (lane >> 4) << 3;
#pragma unroll
    for (int t = 0; t < 4; ++t) {
        const int col = tn0 + t * 16 + row;
        if (col < N) {
            float bv = 0.f;
            if (EPI == EPI_BIAS_GELU || EPI == EPI_BIAS_ADD) bv = bias[col];
#pragma unroll
            for (int mh = 0; mh < 2; ++mh) {
#pragma unroll
                for (int r = 0; r < 8; ++r) {
                    const size_t idx = (size_t)(tm + mh * 16 + rbase + r) * ldc + col;
                    const float v = mh ? dB[t][r] : dA[t][r];
                    if (EPI == EPI_STORE) {
                        C[idx] = v;
                    } else if (EPI == EPI_BIAS_GELU) {
                        const float x = v + bv;   // exact GELU (approximate=False)
                        C[idx] = 0.5f * x * (1.f + erff(x * 0.70710678118654752440f));
                    } else if (EPI == EPI_ACC) {
                        C[idx] += v;
                    } else if (EPI == EPI_BIAS_ADD) {
                        C[idx] = v + bv + R[idx];
                    }
                }
            }
        }
    }
}

// ---------------------------------------------------------------------------
// Per-(b,m) mean / stdev over the L=512 time axis.  x_enc: (B, L, M)
// ---------------------------------------------------------------------------
__global__ __launch_bounds__(256) void norm_stats_kernel(
    const float* __restrict__ x, float* __restrict__ mean, float* __restrict__ stdev)
{
    const int b = blockIdx.x >> 5;
    const int m = blockIdx.x & 31;
    const int t = threadIdx.x;
    __shared__ float s1[256], s2[256];
    float a1 = 0.f, a2 = 0.f;
    for (int n = t; n < SEQ_L; n += 256) {
        const float v = x[((size_t)b * SEQ_L + n) * MVAR + m];
        a1 += v; a2 += v * v;
    }
    s1[t] = a1; s2[t] = a2;
    __syncthreads();
    for (int s = 128; s > 0; s >>= 1) {
        if (t < s) { s1[t] += s1[t + s]; s2[t] += s2[t + s]; }
        __syncthreads();
    }
    if (t == 0) {
        const float mu  = s1[0] * (1.f / SEQ_L);
        const float var = s2[0] * (1.f / SEQ_L) - mu * mu;
        mean[blockIdx.x]  = mu;
        stdev[blockIdx.x] = sqrtf(var + 1e-5f);
    }
}

// Normalized patch gather -> patches[(bm*63+n)*16 + p]
__global__ void gather_patches_kernel(
    const float* __restrict__ x, const float* __restrict__ mean,
    const float* __restrict__ stdev, float* __restrict__ patches)
{
    const int idx = blockIdx.x * 256 + threadIdx.x;
    if (idx >= TOK * P_LEN) return;
    const int p   = idx & 15;
    const int tok = idx >> 4;
    const int n   = tok % N_PATCH;
    const int bm  = tok / N_PATCH;
    const int b   = bm >> 5, m = bm & 31;
    const int t   = n * STRIDE + p;
    const float v = x[((size_t)b * SEQ_L + t) * MVAR + m];
    patches[idx] = (v - mean[bm]) / stdev[bm];
}

// ---------------------------------------------------------------------------
// Causal conv1d (K=4) + SiLU, with optional time flip (dir=1 -> backward pass
// operates on the flipped sequence; output is written in *logical* order).
// xz: (TOK, 512) physical order; channel c in [0,256).  xconv: (TOK, 256).
// ---------------------------------------------------------------------------
__global__ __launch_bounds__(256) void conv_silu_kernel(
    const float* __restrict__ xz, const float* __restrict__ cw,
    const float* __restrict__ cb, float* __restrict__ xconv, int dir)
{
    const int bm = blockIdx.x;
    const int c  = threadIdx.x;
    const float w0 = cw[c * 4 + 0], w1 = cw[c * 4 + 1];
    const float w2 = cw[c * 4 + 2], w3 = cw[c * 4 + 3];
    const float bb = cb[c];
    float x0 = 0.f, x1 = 0.f, x2 = 0.f;
    const float* base = xz + (size_t)bm * N_PATCH * (2 * D_INNER) + c;
    float* out = xconv + (size_t)bm * N_PATCH * D_INNER + c;
    for (int j = 0; j < N_PATCH; ++j) {
        const int ph = dir ? (N_PATCH - 1 - j) : j;
        const float x3 = base[(size_t)ph * (2 * D_INNER)];
        float y = bb + w0 * x0 + w1 * x1 + w2 * x2 + w3 * x3;
        y = y / (1.f + expf(-y));               // SiLU
        out[(size_t)j * D_INNER] = y;
        x0 = x1; x1 = x2; x2 = x3;
    }
}

// ---------------------------------------------------------------------------
// Selective scan, fused with dt-projection+softplus and the output gate.
// One block per sequence (bm); one thread per channel c.
// xdbl  (TOK,40) logical: [0:8]=dt_r, [8:24]=B, [24:40]=C
// xconv (TOK,256) logical = u ; xz (TOK,512) physical (z at col 256+c)
// ygate (TOK,256) written at *physical* rows (un-flips backward direction).
// ---------------------------------------------------------------------------
__global__ __launch_bounds__(256) void scan_kernel(
    const float* __restrict__ xdbl, const float* __restrict__ xconv,
    const float* __restrict__ xz,
    const float* __restrict__ dtpw, const float* __restrict__ dtpb,
    const float* __restrict__ Alog, const float* __restrict__ Dp,
    float* __restrict__ ygate, int dir)
{
    const int bm = blockIdx.x;
    const int c  = threadIdx.x;
    __shared__ float sdt[N_PATCH][DT_RANK];
    __shared__ float sB[N_PATCH][D_STATE];
    __shared__ float sC[N_PATCH][D_STATE];
    for (int i = c; i < N_PATCH * 40; i += 256) {
        const int j = i / 40, e = i % 40;
        const float v = xdbl[((size_t)bm * N_PATCH + j) * 40 + e];
        if (e < 8)        sdt[j][e] = v;
        else if (e < 24)  sB[j][e - 8] = v;
        else              sC[j][e - 24] = v;
    }
    __syncthreads();

    float w[DT_RANK];
#pragma unroll
    for (int r = 0; r < DT_RANK; ++r) w[r] = dtpw[c * DT_RANK + r];
    const float db = dtpb[c];
    float Av[D_STATE];
#pragma unroll
    for (int s = 0; s < D_STATE; ++s) Av[s] = -expf(Alog[c * D_STATE + s]);
    const float Dv = Dp[c];
    float h[D_STATE];
#pragma unroll
    for (int s = 0; s < D_STATE; ++s) h[s] = 0.f;

    for (int j = 0; j < N_PATCH; ++j) {
        float dt = db;
#pragma unroll
        for (int r = 0; r < DT_RANK; ++r) dt += w[r] * sdt[j][r];
        dt = (dt > 20.f) ? dt : log1pf(expf(dt));     // softplus
        const float u   = xconv[((size_t)bm * N_PATCH + j) * D_INNER + c];
        const float dtu = dt * u;
        float y = 0.f;
#pragma unroll
        for (int s = 0; s < D_STATE; ++s) {
            const float dA = expf(dt * Av[s]);
            h[s] = dA * h[s] + dtu * sB[j][s];
            y += h[s] * sC[j][s];
        }
        const int    ph   = dir ? (N_PATCH - 1 - j) : j;
        const size_t prow = (size_t)bm * N_PATCH + ph;
        const float  z    = xz[prow * (2 * D_INNER) + D_INNER + c];
        const float  g    = z / (1.f + expf(-z));     // SiLU(z)
        ygate[prow * D_INNER + c] = (y + Dv * u) * g;
    }
}

// LayerNorm over d=128 per token
__global__ __launch_bounds__(128) void ln_kernel(
    const float* __restrict__ X, const float* __restrict__ g,
    const float* __restrict__ b, float* __restrict__ Y)
{
    const int row = blockIdx.x;
    const int d   = threadIdx.x;
    const float v = X[(size_t)row * D_MODEL + d];
    __shared__ float s1[D_MODEL], s2[D_MODEL];
    s1[d] = v; s2[d] = v * v;
    __syncthreads();
    for (int s = 64; s > 0; s >>= 1) {
        if (d < s) { s1[d] += s1[d + s]; s2[d] += s2[d + s]; }
        __syncthreads();
    }
    const float mu  = s1[0] * (1.f / D_MODEL);
    const float var = s2[0] * (1.f / D_MODEL) - mu * mu;
    const float rs  = rsqrtf(var + 1e-5f);
    Y[(size_t)row * D_MODEL + d] = (v - mu) * rs * g[d] + b[d];
}

// Deterministic split-K reduction for the head, fused with bias + denorm +
// transpose: out[b,p,m] = (sum_c partial[c][bm,p] + head_b[p])*stdev + mean
__global__ void head_reduce_kernel(
    const float* __restrict__ partial, const float* __restrict__ head_b,
    const float* __restrict__ mean, const float* __restrict__ stdev,
    float* __restrict__ out)
{
    const int idx = blockIdx.x * 256 + threadIdx.x;
    if (idx >= BATCH * PRED * MVAR) return;
    const int m  = idx & 31;
    const int p  = (idx >> 5) % PRED;
    const int b  = idx / (PRED * MVAR);
    const int bm = b * MVAR + m;
    float s = head_b[p];
#pragma unroll
    for (int c = 0; c < SPLITK; ++c)
        s += partial[(size_t)c * NBM * PRED + (size_t)bm * PRED + p];
    out[idx] = s * stdev[bm] + mean[bm];
}

// ---------------------------------------------------------------------------
// Host-side launcher
// ---------------------------------------------------------------------------
static void launch_gemm(int epi, const float* A, int lda, const float* W, int ldw,
                        float* C, int ldc, int Mr, int N, int K,
                        const float* bias, const float* R, hipStream_t s)
{
    const int tilesN  = (N + 15) / 16;
    const int groupsN = (tilesN + 3) / 4;
    const int tiles   = (Mr / 32) * groupsN;
    const int blocks  = (tiles + 3) / 4;
    switch (epi) {
        case EPI_STORE:     gemm_wmma_f32<EPI_STORE>    <<<blocks,128,0,s>>>(A,lda,W,ldw,C,ldc,Mr,N,K,bias,R); break;
        case EPI_BIAS_GELU: gemm_wmma_f32<EPI_BIAS_GELU><<<blocks,128,0,s>>>(A,lda,W,ldw,C,ldc,Mr,N,K,bias,R); break;
        case EPI_ACC:       gemm_wmma_f32<EPI_ACC>      <<<blocks,128,0,s>>>(A,lda,W,ldw,C,ldc,Mr,N,K,bias,R); break;
        case EPI_BIAS_ADD:  gemm_wmma_f32<EPI_BIAS_ADD> <<<blocks,128,0,s>>>(A,lda,W,ldw,C,ldc,Mr,N,K,bias,R); break;
    }
}

extern "C" void kernel_launch(void* const* d_in, const int* in_sizes, int n_in,
                              void* d_out, int out_size, void* d_ws, size_t ws_size,
                              hipStream_t stream)
{
    (void)in_sizes; (void)n_in; (void)out_size; (void)ws_size;
    const float* x_enc   = (const float*)d_in[0];
    const float* W_pe    = (const float*)d_in[4];
    const float* in_w    = (const float*)d_in[5];
    const float* conv_w  = (const float*)d_in[6];
    const float* conv_b  = (const float*)d_in[7];
    const float* xproj_w = (const float*)d_in[8];
    const float* dtp_w   = (const float*)d_in[9];
    const float* dtp_b   = (const float*)d_in[10];
    const float* A_log   = (const float*)d_in[11];
    const float* D_p     = (const float*)d_in[12];
    const float* out_w   = (const float*)d_in[13];
    const float* n1_g    = (const float*)d_in[14];
    const float* n1_b    = (const float*)d_in[15];
    const float* f1_w    = (const float*)d_in[16];
    const float* f1_b    = (const float*)d_in[17];
    const float* f2_w    = (const float*)d_in[18];
    const float* f2_b    = (const float*)d_in[19];
    const float* n2_g    = (const float*)d_in[20];
    const float* n2_b    = (const float*)d_in[21];
    const float* ne_g    = (const float*)d_in[22];
    const float* ne_b    = (const float*)d_in[23];
    const float* head_w  = (const float*)d_in[24];
    const float* head_b  = (const float*)d_in[25];
    float* out = (float*)d_out;

    // Workspace layout
    size_t off = 0;
    auto alloc = [&](size_t nbytes) -> float* {
        float* r = (float*)((char*)d_ws + off);
        off += (nbytes + 255) & ~(size_t)255;
        return r;
    };
    float* mean  = alloc((size_t)NBM * 4);
    float* stdev = alloc((size_t)NBM * 4);
    float* h     = alloc((size_t)TOK * D_MODEL * 4);          // h / x1
    float* acc   = alloc((size_t)TOK * D_MODEL * 4);          // residual accumulator
    float* xz    = alloc((size_t)TOK * 2 * D_INNER * 4);      // in-proj out (xi|z); also FFN mid, patches
    float* xconv = alloc((size_t)TOK * D_INNER * 4);          // conv+silu (logical order)
    float* ygate = alloc((size_t)TOK * D_INNER * 4);          // gated scan output; also head partials
    float* xdbl  = alloc((size_t)TOK * 40 * 4);               // x-proj out
    float* patches  = xz;                                     // alias (xz unused at start)
    float* mid      = xz;                                     // alias (xz dead by FFN time)
    float* hpartial = ygate;                                  // alias (ygate dead by head time)

    // 1) instance-norm stats + normalized patch gather + patch embedding
    norm_stats_kernel<<<NBM, 256, 0, stream>>>(x_enc, mean, stdev);
    gather_patches_kernel<<<(TOK * P_LEN + 255) / 256, 256, 0, stream>>>(x_enc, mean, stdev, patches);
    launch_gemm(EPI_STORE, patches, P_LEN, W_pe, P_LEN, h, D_MODEL,
                TOK, D_MODEL, P_LEN, nullptr, nullptr, stream);

    // 2) encoder layers
    for (int l = 0; l < E_LAYERS; ++l) {
        hipMemcpyAsync(acc, h, (size_t)TOK * D_MODEL * 4, hipMemcpyDeviceToDevice, stream);
        for (int dir = 0; dir < 2; ++dir) {
            const int pw = l * 2 + dir;
            launch_gemm(EPI_STORE, h, D_MODEL, in_w + (size_t)pw * 2 * D_INNER * D_MODEL, D_MODEL,
                        xz, 2 * D_INNER, TOK, 2 * D_INNER, D_MODEL, nullptr, nullptr, stream);
            conv_silu_kernel<<<NBM, 256, 0, stream>>>(
                xz, conv_w + (size_t)pw * D_INNER * D_CONV, conv_b + (size_t)pw * D_INNER, xconv, dir);
            launch_gemm(EPI_STORE, xconv, D_INNER, xproj_w + (size_t)pw * 40 * D_INNER, D_INNER,
                        xdbl, 40, TOK, 40, D_INNER, nullptr, nullptr, stream);
            scan_kernel<<<NBM, 256, 0, stream>>>(
                xdbl, xconv, xz,
                dtp_w + (size_t)pw * D_INNER * DT_RANK, dtp_b + (size_t)pw * D_INNER,
                A_log + (size_t)pw * D_INNER * D_STATE, D_p + (size_t)pw * D_INNER,
                ygate, dir);
            launch_gemm(EPI_ACC, ygate, D_INNER, out_w + (size_t)pw * D_MODEL * D_INNER, D_INNER,
                        acc, D_MODEL, TOK, D_MODEL, D_INNER, nullptr, nullptr, stream);
        }
        ln_kernel<<<TOK, D_MODEL, 0, stream>>>(acc, n1_g + l * D_MODEL, n1_b + l * D_MODEL, h); // h := x1
        launch_gemm(EPI_BIAS_GELU, h, D_MODEL, f1_w + (size_t)l * D_FF * D_MODEL, D_MODEL,
                    mid, D_FF, TOK, D_FF, D_MODEL, f1_b + l * D_FF, nullptr, stream);
        launch_gemm(EPI_BIAS_ADD, mid, D_FF, f2_w + (size_t)l * D_MODEL * D_FF, D_FF,
                    acc, D_MODEL, TOK, D_MODEL, D_FF, f2_b + l * D_MODEL, h, stream);
        ln_kernel<<<TOK, D_MODEL, 0, stream>>>(acc, n2_g + l * D_MODEL, n2_b + l * D_MODEL, h);
    }

    // 3) final norm + split-K head + fused reduce/denorm/transpose
    ln_kernel<<<TOK, D_MODEL, 0, stream>>>(h, ne_g, ne_b, acc);
    for (int s = 0; s < SPLITK; ++s) {
        const int k0 = s * HEAD_KC;
        launch_gemm(EPI_STORE, acc + k0, HEAD_K, head_w + k0, HEAD_K,
                    hpartial + (size_t)s * NBM * PRED, PRED,
                    NBM, PRED, HEAD_KC, nullptr, nullptr, stream);
    }
    head_reduce_kernel<<<(BATCH * PRED * MVAR + 255) / 256, 256, 0, stream>>>(
        hpartial, head_b, mean, stdev, out);
}